// EnBaseLayer_7782480740764
// MI455X (gfx1250) — compile-verified
//
#include <hip/hip_runtime.h>
#include <hip/hip_bf16.h>

// EnBaseLayer on MI455X (gfx1250, wave32).
// Block-diagonal feature_joint => edge MLP rows depend on only ONE endpoint,
// so the E=320K edge GEMM collapses to N=20K per-node MLPs + degree count +
// one scatter-add of q1[src] into mi[dst,1,:]. GEMMs use chained
// V_WMMA_F32_16X16X4_F32 (fp32-exact) with 8 independent accumulator tiles
// per wave (full 16x128 output row per k-loop): A pair loaded once per
// k-step, 8 B pairs via one clause of global_load_b64 immediates.

typedef float v2f __attribute__((ext_vector_type(2)));
typedef float v8f __attribute__((ext_vector_type(8)));

#define N_NODES 20000
#define N_EDGES 320000
#define DIM 128
#define LSTR 132  // padded LDS row stride (floats): 132 % 64 = 4 -> no bank conflicts

// One wave computes a full 16x128 f32 row-block: C = A(16x128) * B(128x128) + bias.
// A in LDS (row-major, stride LSTR); Bt = B transposed in global ([j][k], ld=DIM).
// A frag (ISA 7.12.2): lane m=l&15; VGPR0=K even, VGPR1=K odd; hi-half lanes K+2.
// C/D: VGPR r holds row M = r + 8*(l>>4), col N = l&15 (within tile jt).
__device__ __forceinline__ void wmma_gemm_row(const float* sA,
                                              const float* __restrict__ Bt,
                                              const float* __restrict__ bias,
                                              int lane, v8f acc[8]) {
    const int m   = lane & 15;
    const int khi = lane >> 4;  // 0 or 1
    const int n   = lane & 15;
#pragma unroll
    for (int jt = 0; jt < 8; ++jt) {
        float bl = bias[jt * 16 + n];
#pragma unroll
        for (int r = 0; r < 8; ++r) acc[jt][r] = bl;
    }
    const float* ap = sA + m * LSTR + 2 * khi;
    const float* bp = Bt + (size_t)n * DIM + 2 * khi;
    for (int kk = 0; kk < 32; ++kk) {
        v2f a;
        a[0] = ap[4 * kk];
        a[1] = ap[4 * kk + 1];
#pragma unroll
        for (int jt = 0; jt < 8; ++jt) {
            // offset = jt*16*DIM + 4*kk floats -> fits 24-bit imm; one clause of 8 b64
            v2f b = *(const v2f*)(bp + jt * 16 * DIM + 4 * kk);
            acc[jt] = __builtin_amdgcn_wmma_f32_16x16x4_f32(
                false, a, false, b, (short)0, acc[jt], false, false);
        }
    }
}

// 128x128 transpose: dst[j*DIM + k] = src[k*DIM + j]
__global__ void transpose128_kernel(const float* __restrict__ src,
                                    float* __restrict__ dst) {
    int t = blockIdx.x * blockDim.x + threadIdx.x;  // 16384 threads
    int k = t >> 7, j = t & 127;
    dst[j * DIM + k] = src[k * DIM + j];
}

// Per-node edge-MLP precompute: q0[n] = p0*g0, q1[n] = p1*g1.
// 2 waves/block, 16 nodes/wave.
__global__ void __launch_bounds__(64)
node_edge_mlp_kernel(const float* __restrict__ h,
                     const float* __restrict__ wt1e,  // [2][128x128] transposed
                     const float* __restrict__ b1e,
                     const float* __restrict__ wt2e,  // [128x128] transposed
                     const float* __restrict__ b2e,
                     const float* __restrict__ Wg, const float* __restrict__ bg,
                     float* __restrict__ q0, float* __restrict__ q1) {
    __shared__ float sh[2][3][16 * LSTR];
    const int wave  = threadIdx.x >> 5;
    const int lane  = threadIdx.x & 31;
    const int node0 = (blockIdx.x * 2 + wave) * 16;
    float* sA = sh[wave][0];
    float* sT = sh[wave][1];
    float* sP = sh[wave][2];

    for (int i = lane; i < 16 * 32; i += 32) {
        const int m = i >> 5, c = i & 31;
        float4 v = *(const float4*)(h + (size_t)(node0 + m) * DIM + c * 4);
        float* d = sA + m * LSTR + c * 4;
        d[0] = v.x; d[1] = v.y; d[2] = v.z; d[3] = v.w;
    }
    __syncthreads();

    const int n = lane & 15, hi = lane >> 4;
    v8f acc[8];
#pragma unroll 1
    for (int half = 0; half < 2; ++half) {
        // layer 1 + ReLU -> sT
        wmma_gemm_row(sA, wt1e + (size_t)half * DIM * DIM, b1e, lane, acc);
#pragma unroll
        for (int jt = 0; jt < 8; ++jt)
#pragma unroll
            for (int r = 0; r < 8; ++r) {
                float t = acc[jt][r]; t = t > 0.f ? t : 0.f;
                sT[(r + 8 * hi) * LSTR + jt * 16 + n] = t;
            }
        __syncthreads();
        // layer 2 -> sP
        wmma_gemm_row(sT, wt2e, b2e, lane, acc);
#pragma unroll
        for (int jt = 0; jt < 8; ++jt)
#pragma unroll
            for (int r = 0; r < 8; ++r)
                sP[(r + 8 * hi) * LSTR + jt * 16 + n] = acc[jt][r];
        __syncthreads();
        // gate: g = sigmoid(p . Wg + bg); 2 lanes per row, 64 elems each
        const int row = lane & 15, seg = lane >> 4;
        float s = 0.f;
        for (int j = 0; j < 64; ++j)
            s += sP[row * LSTR + seg * 64 + j] * Wg[seg * 64 + j];
        s += __shfl_xor(s, 16);
        const float g = 1.f / (1.f + expf(-(s + bg[0])));
        float* q = (half == 0) ? q0 : q1;
        for (int j = 0; j < 64; ++j)
            q[(size_t)(node0 + row) * DIM + seg * 64 + j] =
                sP[row * LSTR + seg * 64 + j] * g;
        __syncthreads();
    }
}

__global__ void init_ws_kernel(float* __restrict__ miB, int* __restrict__ indeg) {
    int t = blockIdx.x * blockDim.x + threadIdx.x;
    if (t < N_NODES * DIM) miB[t] = 0.f;
    if (t < N_NODES) indeg[t] = 0;
}

__global__ void degree_kernel(const int* __restrict__ ei, int* __restrict__ indeg) {
    int e = blockIdx.x * blockDim.x + threadIdx.x;
    if (e < N_EDGES) atomicAdd(&indeg[ei[N_EDGES + e]], 1);
}

// mi[:,1,:]: one wave per edge; float4 load + 4 fp32 atomics per lane (L2-resident).
__global__ void scatter_kernel(const int* __restrict__ ei,
                               const float* __restrict__ q1,
                               float* __restrict__ miB) {
    int t = blockIdx.x * blockDim.x + threadIdx.x;  // E*32 threads
    int e = t >> 5, lane = t & 31;
    if (e >= N_EDGES) return;
    int src = ei[e];
    int dst = ei[N_EDGES + e];
    float4 v = *(const float4*)(q1 + (size_t)src * DIM + lane * 4);
    float* d = miB + (size_t)dst * DIM + lane * 4;
    atomicAdd(d + 0, v.x);
    atomicAdd(d + 1, v.y);
    atomicAdd(d + 2, v.z);
    atomicAdd(d + 3, v.w);
}

// h_new[n, rs, :] = h[n] + mlp2 over rows {indeg*q0, miB, h}, W1n top/top/bottom.
__global__ void __launch_bounds__(64)
node_update_kernel(const float* __restrict__ h,
                   const float* __restrict__ q0,
                   const float* __restrict__ miB,
                   const int* __restrict__ indeg,
                   const float* __restrict__ wt1n,  // [2][128x128] transposed
                   const float* __restrict__ b1n,
                   const float* __restrict__ wt2n,  // [128x128] transposed
                   const float* __restrict__ b2n,
                   float* __restrict__ out) {
    __shared__ float sh[2][3][16 * LSTR];
    const int wave  = threadIdx.x >> 5;
    const int lane  = threadIdx.x & 31;
    const int node0 = (blockIdx.x * 2 + wave) * 16;
    float* sH = sh[wave][0];
    float* sA = sh[wave][1];
    float* sT = sh[wave][2];

    for (int i = lane; i < 16 * 32; i += 32) {
        int m = i >> 5, c = i & 31;
        float4 v = *(const float4*)(h + (size_t)(node0 + m) * DIM + c * 4);
        float* d = sH + m * LSTR + c * 4;
        d[0] = v.x; d[1] = v.y; d[2] = v.z; d[3] = v.w;
    }
    __syncthreads();

    const int n = lane & 15, hi = lane >> 4;
    v8f acc[8];
#pragma unroll 1
    for (int rs = 0; rs < 3; ++rs) {
        const float* S = (rs == 0) ? q0 : (rs == 1 ? miB : h);
        for (int i = lane; i < 16 * 32; i += 32) {
            int m = i >> 5, c = i & 31;
            int node = node0 + m;
            float4 v = *(const float4*)(S + (size_t)node * DIM + c * 4);
            float sc = (rs == 0) ? (float)indeg[node] : 1.f;
            float* d = sA + m * LSTR + c * 4;
            d[0] = v.x * sc; d[1] = v.y * sc; d[2] = v.z * sc; d[3] = v.w * sc;
        }
        __syncthreads();
        wmma_gemm_row(sA, wt1n + (size_t)(rs == 2 ? DIM * DIM : 0), b1n, lane, acc);
#pragma unroll
        for (int jt = 0; jt < 8; ++jt)
#pragma unroll
            for (int r = 0; r < 8; ++r) {
                float t = acc[jt][r]; t = t > 0.f ? t : 0.f;
                sT[(r + 8 * hi) * LSTR + jt * 16 + n] = t;
            }
        __syncthreads();
        wmma_gemm_row(sT, wt2n, b2n, lane, acc);
#pragma unroll
        for (int jt = 0; jt < 8; ++jt)
#pragma unroll
            for (int r = 0; r < 8; ++r) {
                int m = r + 8 * hi;
                float y = acc[jt][r] + sH[m * LSTR + jt * 16 + n];
                out[((size_t)(node0 + m) * 3 + rs) * DIM + jt * 16 + n] = y;
            }
        __syncthreads();
    }
}

__global__ void copy_x_kernel(const float* __restrict__ x, float* __restrict__ out) {
    int t = blockIdx.x * blockDim.x + threadIdx.x;
    if (t < N_NODES * 3) out[(size_t)N_NODES * 3 * DIM + t] = x[t];
}

extern "C" void kernel_launch(void* const* d_in, const int* in_sizes, int n_in,
                              void* d_out, int out_size, void* d_ws, size_t ws_size,
                              hipStream_t stream) {
    const float* h   = (const float*)d_in[0];
    const float* x   = (const float*)d_in[1];
    const int*   ei  = (const int*)d_in[2];
    const float* W1e = (const float*)d_in[3];
    const float* b1e = (const float*)d_in[4];
    const float* W2e = (const float*)d_in[5];
    const float* b2e = (const float*)d_in[6];
    const float* Wg  = (const float*)d_in[7];
    const float* bg  = (const float*)d_in[8];
    const float* W1n = (const float*)d_in[9];
    const float* b1n = (const float*)d_in[10];
    const float* W2n = (const float*)d_in[11];
    const float* b2n = (const float*)d_in[12];
    float* out = (float*)d_out;

    const size_t MAT = (size_t)DIM * DIM;  // 16384 floats
    char* ws = (char*)d_ws;
    float* q0    = (float*)ws; ws += (size_t)N_NODES * DIM * sizeof(float);
    float* q1    = (float*)ws; ws += (size_t)N_NODES * DIM * sizeof(float);
    float* miB   = (float*)ws; ws += (size_t)N_NODES * DIM * sizeof(float);
    int*   indeg = (int*)ws;   ws += (size_t)N_NODES * sizeof(int);
    float* wt1e  = (float*)ws; ws += 2 * MAT * sizeof(float);
    float* wt2e  = (float*)ws; ws += MAT * sizeof(float);
    float* wt1n  = (float*)ws; ws += 2 * MAT * sizeof(float);
    float* wt2n  = (float*)ws;

    // weight transposes (one-time per launch; L2-resident afterwards)
    transpose128_kernel<<<64, 256, 0, stream>>>(W1e,        wt1e);
    transpose128_kernel<<<64, 256, 0, stream>>>(W1e + MAT,  wt1e + MAT);
    transpose128_kernel<<<64, 256, 0, stream>>>(W2e,        wt2e);
    transpose128_kernel<<<64, 256, 0, stream>>>(W1n,        wt1n);
    transpose128_kernel<<<64, 256, 0, stream>>>(W1n + MAT,  wt1n + MAT);
    transpose128_kernel<<<64, 256, 0, stream>>>(W2n,        wt2n);

    init_ws_kernel<<<(N_NODES * DIM + 255) / 256, 256, 0, stream>>>(miB, indeg);
    node_edge_mlp_kernel<<<N_NODES / 32, 64, 0, stream>>>(h, wt1e, b1e, wt2e, b2e,
                                                          Wg, bg, q0, q1);
    degree_kernel<<<(N_EDGES + 255) / 256, 256, 0, stream>>>(ei, indeg);
    scatter_kernel<<<(N_EDGES * 32) / 256, 256, 0, stream>>>(ei, q1, miB);
    node_update_kernel<<<N_NODES / 32, 64, 0, stream>>>(h, q0, miB, indeg,
                                                        wt1n, b1n, wt2n, b2n, out);
    copy_x_kernel<<<(N_NODES * 3 + 255) / 256, 256, 0, stream>>>(x, out);
}